// AdaptiveGaussianConv_64415919506209
// MI455X (gfx1250) — compile-verified
//
#include <hip/hip_runtime.h>
#include <hip/hip_bf16.h>
#include <math.h>

// Problem constants (fixed by setup_inputs)
#define BATCH 8
#define CHAN 64
#define HH 384
#define WW 384
#define PLANE (HH * WW)          // 147456
#define NPLANES (BATCH * CHAN)   // 512
#define KS 7
#define HALO 3
#define BAND 16                  // output rows per conv block
#define NBANDS (HH / BAND)       // 24

typedef __attribute__((ext_vector_type(2))) float v2f;
typedef __attribute__((ext_vector_type(8))) float v8f;

// ---------------------------------------------------------------------------
// Kernel A: global average pool.  512 blocks (one per (b,c) plane) x 256 thr.
// 302 MB read -> ~13 us at 23.3 TB/s.  float4 loads, LDS tree reduce.
// ---------------------------------------------------------------------------
__global__ __launch_bounds__(256) void pool_kernel(const float* __restrict__ x,
                                                   float* __restrict__ pooled) {
    const int bc = blockIdx.x;
    const float4* p = (const float4*)(x + (size_t)bc * PLANE);
    const int n4 = PLANE / 4;  // 36864
    float s = 0.0f;
    for (int i = threadIdx.x; i < n4; i += 256) {
        if (i + 1024 < n4) __builtin_prefetch(&p[i + 1024], 0, 1);
        float4 v = p[i];
        s += (v.x + v.y) + (v.z + v.w);
    }
    __shared__ float red[256];
    red[threadIdx.x] = s;
    __syncthreads();
    for (int off = 128; off > 0; off >>= 1) {
        if (threadIdx.x < off) red[threadIdx.x] += red[threadIdx.x + off];
        __syncthreads();
    }
    if (threadIdx.x == 0) pooled[bc] = red[0] * (1.0f / (float)PLANE);
}

// ---------------------------------------------------------------------------
// Kernel B: param predictor.  One wave32 (EXEC all ones — required for WMMA).
// h_pre[16,16] = pooled_pad[16,64] @ w1^T  via 16 chained V_WMMA_F32_16X16X4_F32.
// A layout (ISA 7.12.2, 32-bit A 16x4): lanes 0-15 -> K0(v0)/K1(v1),
// lanes 16-31 -> K2(v0)/K3(v1).  B rows striped across lane halves per VGPR.
// D layout: VGPR i -> row i (lanes 0-15) / row i+8 (lanes 16-31).
// Then SiLU, 16->3 dense, softplus/tanh, and the two normalized 1-D 7-tap
// Gaussian factors per batch (separable; (sum wx)(sum wy) = full 7x7 sum).
// ---------------------------------------------------------------------------
__global__ __launch_bounds__(32) void mlp_kernel(const float* __restrict__ pooled,
                                                 const float* __restrict__ w1,
                                                 const float* __restrict__ b1,
                                                 const float* __restrict__ w2,
                                                 const float* __restrict__ b2,
                                                 float* __restrict__ kern) {
    const int lane = threadIdx.x;       // 0..31
    const int m = lane & 15;            // row (batch, padded to 16) / col n for B
    const int kh = lane >> 4;           // lane-half selector
    __shared__ float s_h[BATCH * 16];

#if __has_builtin(__builtin_amdgcn_wmma_f32_16x16x4_f32)
    v8f acc = {0.f, 0.f, 0.f, 0.f, 0.f, 0.f, 0.f, 0.f};
#pragma unroll
    for (int s = 0; s < 16; ++s) {
        const int k0 = s * 4;
        v2f a, b;
        if (m < BATCH) {
            a.x = pooled[m * CHAN + k0 + 2 * kh];
            a.y = pooled[m * CHAN + k0 + 2 * kh + 1];
        } else {
            a.x = 0.f; a.y = 0.f;
        }
        // B[k][n] = w1[n][k]; n = lane&15.  v0 holds rows {k0,k0+1}, v1 {k0+2,k0+3}.
        b.x = w1[m * CHAN + k0 + kh];
        b.y = w1[m * CHAN + k0 + 2 + kh];
        acc = __builtin_amdgcn_wmma_f32_16x16x4_f32(
            /*neg_a=*/false, a, /*neg_b=*/false, b,
            /*c_mod=*/(short)0, acc, /*reuse_a=*/false, /*reuse_b=*/false);
    }
    if (lane < 16) {  // lanes hold D rows 0..7 = batches 0..7, column n = lane
#pragma unroll
        for (int i = 0; i < BATCH; ++i) {
            float z = acc[i] + b1[lane];
            s_h[i * 16 + lane] = z / (1.0f + expf(-z));   // SiLU
        }
    }
#else
    if (lane < BATCH) {
        for (int n = 0; n < 16; ++n) {
            float z = b1[n];
            for (int c = 0; c < CHAN; ++c) z += pooled[lane * CHAN + c] * w1[n * CHAN + c];
            s_h[lane * 16 + n] = z / (1.0f + expf(-z));
        }
    }
#endif
    __syncthreads();

    if (lane < BATCH) {
        float p[3];
#pragma unroll
        for (int j = 0; j < 3; ++j) {
            float z = b2[j];
#pragma unroll
            for (int n = 0; n < 16; ++n) z += s_h[lane * 16 + n] * w2[j * 16 + n];
            p[j] = z;
        }
        // softplus (stable), tanh offsets
        float sigma = fmaxf(p[0], 0.0f) + log1pf(expf(-fabsf(p[0])));
        float cx = 3.0f + 2.0f * tanhf(p[1]);
        float cy = 3.0f + 2.0f * tanhf(p[2]);
        float inv2s2 = 1.0f / (2.0f * sigma * sigma);
        float wx[KS], wy[KS], sx = 0.f, sy = 0.f;
#pragma unroll
        for (int t = 0; t < KS; ++t) {
            float ddx = (float)t - cx, ddy = (float)t - cy;
            wx[t] = expf(-ddx * ddx * inv2s2);
            wy[t] = expf(-ddy * ddy * inv2s2);
            sx += wx[t];
            sy += wy[t];
        }
        float rx = 1.0f / sx, ry = 1.0f / sy;
#pragma unroll
        for (int t = 0; t < KS; ++t) {
            kern[lane * 16 + t]     = wx[t] * rx;   // horizontal taps
            kern[lane * 16 + 8 + t] = wy[t] * ry;   // vertical taps
        }
    }
}

// ---------------------------------------------------------------------------
// Kernel C: separable depthwise conv, one 16-row band of one plane per block.
// Load 22x384 rows (zero halo) -> LDS; vertical 7-tap -> padded LDS temp
// (16 x [4 zero | 384 | 4 zero]); horizontal 7-tap from 3 b128 LDS reads per
// float4 output; coalesced b128 global store.  Memory-bound: ~26 us total.
// ---------------------------------------------------------------------------
__global__ __launch_bounds__(256) void conv_kernel(const float* __restrict__ x,
                                                   const float* __restrict__ kern,
                                                   float* __restrict__ out) {
    const int band = blockIdx.x;   // 0..23
    const int bc   = blockIdx.y;   // 0..511
    const int b    = bc >> 6;
    const int tid  = threadIdx.x;

    __shared__ float s_in[(BAND + 2 * HALO) * WW];   // 22*384 floats = 33 KB
    __shared__ float s_tmp[BAND * (WW + 8)];         // 16*392 floats = 24.5 KB

    const float* kb = kern + b * 16;
    float wx[KS], wy[KS];
#pragma unroll
    for (int t = 0; t < KS; ++t) { wx[t] = kb[t]; wy[t] = kb[8 + t]; }

    const size_t plane = (size_t)bc * PLANE;
    const float* src = x + plane;
    float* dst = out + plane;
    const int row0 = band * BAND - HALO;

    // zero the 4-float pads of s_tmp (left/right of each of 16 rows)
    if (tid < 32) {
        int r = tid >> 1, side = tid & 1;
        float4* pp = (float4*)&s_tmp[r * (WW + 8) + side * (WW + 4)];
        *pp = make_float4(0.f, 0.f, 0.f, 0.f);
    }

    // load 22 rows x 96 float4 with zero row-halo
    for (int i = tid; i < (BAND + 2 * HALO) * (WW / 4); i += 256) {
        int r = i / (WW / 4), c4 = i % (WW / 4);
        int gr = row0 + r;
        float4 v;
        if (gr >= 0 && gr < HH) v = ((const float4*)(src + (size_t)gr * WW))[c4];
        else                    v = make_float4(0.f, 0.f, 0.f, 0.f);
        ((float4*)&s_in[r * WW])[c4] = v;
    }
    __syncthreads();

    // vertical 7-tap: temp[ty][x] = sum_t wy[t] * in[ty+t][x]
    for (int i = tid; i < BAND * (WW / 4); i += 256) {
        int ty = i / (WW / 4), c4 = i % (WW / 4);
        float4 acc = make_float4(0.f, 0.f, 0.f, 0.f);
#pragma unroll
        for (int t = 0; t < KS; ++t) {
            float4 v = ((const float4*)&s_in[(ty + t) * WW])[c4];
            float w = wy[t];
            acc.x += w * v.x; acc.y += w * v.y; acc.z += w * v.z; acc.w += w * v.w;
        }
        ((float4*)&s_tmp[ty * (WW + 8) + 4])[c4] = acc;
    }
    __syncthreads();

    // horizontal 7-tap + store: out[x+k] = sum_t wx[t] * temp_padded[x+k+1+t]
    for (int i = tid; i < BAND * (WW / 4); i += 256) {
        int ty = i / (WW / 4), c4 = i % (WW / 4);
        const float* p = &s_tmp[ty * (WW + 8) + c4 * 4];  // = unpadded col c4*4 - 4
        float4 a0 = ((const float4*)p)[0];
        float4 a1 = ((const float4*)p)[1];
        float4 a2 = ((const float4*)p)[2];
        float f[12] = {a0.x, a0.y, a0.z, a0.w, a1.x, a1.y, a1.z, a1.w,
                       a2.x, a2.y, a2.z, a2.w};
        float o[4];
#pragma unroll
        for (int k = 0; k < 4; ++k) {
            float s = 0.f;
#pragma unroll
            for (int t = 0; t < KS; ++t) s += wx[t] * f[k + 1 + t];
            o[k] = s;
        }
        ((float4*)(dst + (size_t)(band * BAND + ty) * WW))[c4] =
            make_float4(o[0], o[1], o[2], o[3]);
    }
}

// ---------------------------------------------------------------------------
extern "C" void kernel_launch(void* const* d_in, const int* in_sizes, int n_in,
                              void* d_out, int out_size, void* d_ws, size_t ws_size,
                              hipStream_t stream) {
    const float* x  = (const float*)d_in[0];
    const float* w1 = (const float*)d_in[1];
    const float* b1 = (const float*)d_in[2];
    const float* w2 = (const float*)d_in[3];
    const float* b2 = (const float*)d_in[4];
    float* out = (float*)d_out;

    float* pooled = (float*)d_ws;        // 512 floats
    float* kern   = pooled + NPLANES;    // 8 * 16 floats (wx[0..6], wy @ +8)

    pool_kernel<<<NPLANES, 256, 0, stream>>>(x, pooled);
    mlp_kernel<<<1, 32, 0, stream>>>(pooled, w1, b1, w2, b2, kern);
    conv_kernel<<<dim3(NBANDS, NPLANES), 256, 0, stream>>>(x, kern, out);
}